// TransGQAIndexer_46480136077920
// MI455X (gfx1250) — compile-verified
//
#include <hip/hip_runtime.h>
#include <math.h>

// MI455X / gfx1250 (wave32, WMMA). Full-f32 pipeline: workload is HBM-bound
// (~390MB @ 23.3TB/s => ~17us floor, ~11 GFLOP matrix work), so we keep f32
// end-to-end via V_WMMA_F32_16X16X4_F32.
//
// Restructure: fold + band-proj + nope-proj are composed into two precomputed
// 1024x64 matrices R (rope) and M (nope):
//   index_q[t,grp,:] = [ qrow(q)@R ; qrow(q_orig)@M ]   (rows = 32768)
//   index_k[t,:]     = [ k[t]@R    ; k_orig[t]@M    ]   (rows = 8192)
//   weights[t,grp]   = || v-head projections ||          (VALU)

typedef __attribute__((ext_vector_type(2))) float v2f;
typedef __attribute__((ext_vector_type(8))) float v8f;

// ---------------------------------------------------------------------------
// Kernel A: build composed matrices R[1024][64], M[1024][64] in workspace.
// in-coord c = g*128 + d ; d -> (part, band, col_in) via the fold permutation.
// ---------------------------------------------------------------------------
__global__ __launch_bounds__(256) void build_rm_kernel(
    const float* __restrict__ qkw,  // [32][16][16]
    const float* __restrict__ wn,   // [64][960]
    float* __restrict__ R,          // [1024][64]
    float* __restrict__ M) {        // [1024][64]
  int c = blockIdx.x * 256 + threadIdx.x;
  if (c >= 1024) return;
  int g = c >> 7, d = c & 127;
  int part = d >> 6, dp = d & 63;
  int n = dp >> 1, f = dp & 1;
  int band = f * 16 + (n >> 1);
  int col_in = (n & 1) * 8 + g;
  const float* wrow = qkw + band * 256 + col_in * 16;  // qkw[band][col_in][*]

  int jr = part * 32 + band;
  for (int j = 0; j < 64; ++j) R[c * 64 + j] = (j == jr) ? wrow[0] : 0.0f;

  for (int j = 0; j < 64; ++j) {
    const float* wnrow = wn + j * 960 + part * 480 + band * 15;
    float s = 0.0f;
#pragma unroll
    for (int col = 1; col < 16; ++col) s += wrow[col] * wnrow[col - 1];
    M[c * 64 + j] = s;
  }
}

// ---------------------------------------------------------------------------
// Kernel B: GEMM  out[rows][128] (cols [col_off,col_off+64)) = A[rows][1024]*B
// MODE 0: A row r contiguous (k / k_orig).
// MODE 1: row r=(t,grp): element c=(g,d) at  t*4096 + grp*128 + g*512 + d
//         -> two 128-column blocks per 256-K chunk, each linear (stride 512).
// One wave = 16 rows x 64 cols, K=1024 via V_WMMA_F32_16X16X4_F32.
// B staged in LDS K-pair-interleaved ([k/2][n][2]) so each B fragment is a
// single aligned ds_load_b64.
// ---------------------------------------------------------------------------
template <int MODE>
__global__ __launch_bounds__(256) void gemm_idx_kernel(
    const float* __restrict__ A, const float* __restrict__ B,
    float* __restrict__ out, int col_off) {
  __shared__ float ldsB[16384];  // 64 KB = 128 K-pairs x 64 cols x 2

  const int tid = threadIdx.x;
  const int lane = tid & 31;
  const int wave = tid >> 5;
  const int m = lane & 15;      // A row within tile == output column index n
  const int khalf = lane >> 4;  // selects K pair {0,1} vs {2,3}
  const int row0 = blockIdx.x * 128 + wave * 16;
  const int r = row0 + m;

  long abase;
  if (MODE == 0)
    abase = (long)r * 1024;
  else
    abase = (long)(r >> 2) * 4096 + (long)(r & 3) * 128;

  v8f acc[4];
#pragma unroll
  for (int nt = 0; nt < 4; ++nt) acc[nt] = (v8f){0, 0, 0, 0, 0, 0, 0, 0};

  for (int kc = 0; kc < 4; ++kc) {
    __syncthreads();
    // stage B rows [kc*256, kc*256+256) pair-interleaved: lds[p][n][2]
    const float* bsrc = B + kc * 16384;
    for (int p = tid; p < 8192; p += 256) {
      int kpair = p >> 6, n = p & 63;
      v2f val;
      val.x = bsrc[kpair * 128 + n];
      val.y = bsrc[kpair * 128 + 64 + n];
      *(v2f*)(ldsB + p * 2) = val;
    }
    __syncthreads();

    // two 128-column sub-blocks per chunk (for MODE 1 each is stride-512)
#pragma unroll
    for (int gblk = 0; gblk < 2; ++gblk) {
      const float* ap =
          A + abase + 2 * khalf +
          (MODE == 0 ? (long)(kc * 256 + gblk * 128) : (long)(kc * 2 + gblk) * 512);
      __builtin_prefetch(ap + 64, 0, 1);  // global_prefetch_b8
      const int pbase = (gblk * 128) / 2 + khalf;  // K-pair index base
#pragma unroll 2
      for (int kk = 0; kk < 128; kk += 4) {
        v2f a = *(const v2f*)(ap + kk);  // A[m][k], A[m][k+1]
        const float* bp = ldsB + (pbase + (kk >> 1)) * 128 + m * 2;
#pragma unroll
        for (int nt = 0; nt < 4; ++nt) {
          v2f b = *(const v2f*)(bp + nt * 32);  // ds_load_b64: B[k][n],B[k+1][n]
          acc[nt] = __builtin_amdgcn_wmma_f32_16x16x4_f32(
              false, a, false, b, (short)0, acc[nt], false, false);
        }
      }
    }
  }

  // C/D layout: lane holds rows (ci + 8*khalf), col n = m
#pragma unroll
  for (int nt = 0; nt < 4; ++nt) {
#pragma unroll
    for (int ci = 0; ci < 8; ++ci) {
      int row_out = row0 + ci + 8 * khalf;
      out[(long)row_out * 128 + col_off + nt * 16 + m] = acc[nt][ci];
    }
  }
}

// ---------------------------------------------------------------------------
// Kernel C: weights[t][grp] = || w[t, h=g*4+grp, rr] over (g, rr) ||
// One wave per token; lane l owns head h=l (4 r-dots of length 128).
// Cross-lane reduce over lanes with equal l%4 via shfl_xor (wave32).
// ---------------------------------------------------------------------------
__global__ __launch_bounds__(256) void weights_kernel(
    const float* __restrict__ v,   // [N][8][128]
    const float* __restrict__ vt,  // [32][128][4]
    float* __restrict__ wout) {    // [N][4]
  const int lane = threadIdx.x & 31;
  const int wv = threadIdx.x >> 5;
  const long t = (long)blockIdx.x * 8 + wv;

  const float* vrow = v + t * 1024 + (lane >> 2) * 128;  // v[t, h/4, :]
  const float* vtrow = vt + lane * 512;                  // vt[h, :, :]
  float4 acc = {0.f, 0.f, 0.f, 0.f};
  for (int d = 0; d < 128; ++d) {
    float vv = vrow[d];
    float4 w4 = *(const float4*)(vtrow + d * 4);
    acc.x += vv * w4.x; acc.y += vv * w4.y;
    acc.z += vv * w4.z; acc.w += vv * w4.w;
  }
  float s = acc.x * acc.x + acc.y * acc.y + acc.z * acc.z + acc.w * acc.w;
  s += __shfl_xor(s, 4);
  s += __shfl_xor(s, 8);
  s += __shfl_xor(s, 16);
  if (lane < 4) wout[t * 4 + lane] = sqrtf(s);
}

// ---------------------------------------------------------------------------
extern "C" void kernel_launch(void* const* d_in, const int* in_sizes, int n_in,
                              void* d_out, int out_size, void* d_ws,
                              size_t ws_size, hipStream_t stream) {
  const float* q      = (const float*)d_in[0];  // [8192][32][128]
  const float* k      = (const float*)d_in[1];  // [8192][8][128]
  const float* v      = (const float*)d_in[2];  // [8192][8][128]
  const float* q_orig = (const float*)d_in[3];
  const float* k_orig = (const float*)d_in[4];
  const float* qkw    = (const float*)d_in[5];  // [32][16][16]
  const float* wn     = (const float*)d_in[6];  // [64][960]
  const float* vt     = (const float*)d_in[7];  // [32][128][4]

  float* out = (float*)d_out;
  float* iq = out;            // index_q : 8192*4*128
  float* ik = out + 4194304;  // index_k : 8192*128
  float* wt = out + 5242880;  // weights : 8192*4

  float* R = (float*)d_ws;    // [1024][64]
  float* M = R + 65536;       // [1024][64]  (512 KB total workspace)

  build_rm_kernel<<<4, 256, 0, stream>>>(qkw, wn, R, M);

  // q rope / q nope (rows = 32768 -> 256 blocks of 128 rows)
  gemm_idx_kernel<1><<<256, 256, 0, stream>>>(q,      R, iq, 0);
  gemm_idx_kernel<1><<<256, 256, 0, stream>>>(q_orig, M, iq, 64);
  // k rope / k nope (rows = 8192 -> 64 blocks)
  gemm_idx_kernel<0><<<64, 256, 0, stream>>>(k,      R, ik, 0);
  gemm_idx_kernel<0><<<64, 256, 0, stream>>>(k_orig, M, ik, 64);

  weights_kernel<<<1024, 256, 0, stream>>>(v, vt, wt);
}